// CNN_77077483094746
// MI455X (gfx1250) — compile-verified
//
#include <hip/hip_runtime.h>

typedef __attribute__((ext_vector_type(2))) float v2f;
typedef __attribute__((ext_vector_type(8))) float v8f;

#define OFC 118
#define KPAD 120
#define TDN 21
#define XBUF_SZ 2552   // 18*140 = 2520 + 32 slack (in-bounds don't-care reads)

__device__ __forceinline__ v8f wmma4(v2f a, v2f b, v8f c) {
  // V_WMMA_F32_16X16X4_F32 : D = A(16x4) * B(4x16) + C
  return __builtin_amdgcn_wmma_f32_16x16x4_f32(false, a, false, b, (short)0, c, false, false);
}

__global__ __launch_bounds__(256) void fused_net(
    const float* __restrict__ X,
    const float* __restrict__ tdA_in_w, const float* __restrict__ tdA_in_b,
    const float* __restrict__ tdA_out_w, const float* __restrict__ tdA_out_b,
    const float* __restrict__ tdB_in_w, const float* __restrict__ tdB_in_b,
    const float* __restrict__ tdB_out_w, const float* __restrict__ tdB_out_b,
    const float* __restrict__ cm_in_w, const float* __restrict__ cm_in_b,
    const float* __restrict__ cm_out_w, const float* __restrict__ cm_out_b,
    const float* __restrict__ projA, const float* __restrict__ projB,
    const float* __restrict__ conv_w, const float* __restrict__ conv_b,
    const float* __restrict__ fc1_w, const float* __restrict__ fc1_b,
    const float* __restrict__ fc2_w, const float* __restrict__ fc2_b,
    float* __restrict__ out)
{
  __shared__ union {
    struct {                 // temporal-distance MHA scratch (reused for A then B)
      float kp[32][128];     // keys proj   (21 valid rows, pads zero)
      float vp[32][128];     // values proj
      float qp[16][128];     // query proj, later reused as attn@V
      float att[16][128];    // MHA output
      float sc[16][32];      // 16x21 scores (pads zero)
    } td;
    struct {                 // cross-modal branch scratch
      float qp[128][16];     // 118 valid rows, pads zero
      float kp[128][16];
      float vp[128][16];
      float o [128][16];     // branch MHA output
      float sc[16][128];     // one 16-row score strip; later reused as conv y-buffer
      float tmp[16][16];     // C-frag -> A-frag round trip
    } br;
  } U;
  __shared__ float xbuf[XBUF_SZ];   // staged input tensor, row stride 140
  __shared__ float attrowA[128], attrowB[128];
  __shared__ float gsum[128];
  __shared__ float sval[16];
  __shared__ float feats[40];
  __shared__ float hbuf[40];
  __shared__ int iSel;

  const int tid  = threadIdx.x;
  const int wave = tid >> 5;
  const int lane = tid & 31;
  const int half = lane >> 4;   // K-pair / row-half select per WMMA layout
  const int lm   = lane & 15;   // M (A-frag) or N (B-frag) index
  const int koff = half * 2;

  { // zero LDS so padded rows/cols contribute exactly 0 to WMMA
    float* z = (float*)&U;
    const int n4 = (int)(sizeof(U) / 4);
    for (int i = tid; i < n4; i += 256) z[i] = 0.0f;
    for (int i = 2520 + tid; i < XBUF_SZ; i += 256) xbuf[i] = 0.0f;
    if (tid < 128) { attrowA[tid] = 0.f; attrowB[tid] = 0.f; }
    if (tid < 40) { feats[tid] = 0.f; hbuf[tid] = 0.f; }
  }
  __syncthreads();

  // ---- stage x into LDS: Tensor Data Mover when available (one DMA) ----
#if __has_builtin(__builtin_amdgcn_tensor_load_to_lds) && __has_builtin(__builtin_amdgcn_s_wait_tensorcnt)
  if (wave == 0) {
    typedef unsigned int v4u __attribute__((ext_vector_type(4)));
    typedef int v4i __attribute__((ext_vector_type(4)));
    typedef int v8i __attribute__((ext_vector_type(8)));
    const unsigned long long ga = (unsigned long long)(uintptr_t)X;
    const unsigned int lds = (unsigned int)(uintptr_t)&xbuf[0];
    v4u g0;
    g0[0] = 1u;                                              // count=1 (valid user D#)
    g0[1] = lds;                                             // lds_addr
    g0[2] = (unsigned int)(ga & 0xffffffffu);                // global_addr[31:0]
    g0[3] = (unsigned int)((ga >> 32) & 0x01ffffffu) | (2u << 30); // addr[56:32] | type=2
    v8i g1;
    g1[0] = (2 << 16);                        // data_size = 4B
    g1[1] = (int)(140u << 16);                // tensor_dim0 = 140 (bits 79:48, low half)
    g1[2] = (int)(18u << 16);                 // tensor_dim1 = 18  (bits 111:80, low half)
    g1[3] = (int)(140u << 16);                // tile_dim0 = 140   (bits 127:112)
    g1[4] = 18;                               // tile_dim1 = 18, tile_dim2 = 0
    g1[5] = 140;                              // tensor_dim0_stride = 140
    g1[6] = 0; g1[7] = 0;
    v4i gz = {0, 0, 0, 0};
#if __clang_major__ >= 23
    v8i gz8 = {0, 0, 0, 0, 0, 0, 0, 0};
    __builtin_amdgcn_tensor_load_to_lds(g0, g1, gz, gz, gz8, 0);
#else
    __builtin_amdgcn_tensor_load_to_lds(g0, g1, gz, gz, 0);
#endif
    __builtin_amdgcn_s_wait_tensorcnt(0);
  }
#else
  for (int i = tid; i < 2520; i += 256) xbuf[i] = X[i];
#endif
  __syncthreads();

  const float RSQ = 0.09205746178983235f; // 1/sqrt(118)

  // ================= temporal-distance MHA (A then B) =================
  for (int sel = 0; sel < 2; ++sel) {
    const float* in_w = sel ? tdB_in_w  : tdA_in_w;
    const float* in_b = sel ? tdB_in_b  : tdA_in_b;
    const float* ow   = sel ? tdB_out_w : tdA_out_w;
    const float* ob   = sel ? tdB_out_b : tdA_out_b;
    const int xrow    = sel ? 17 : 0;

    // qp (8 tiles) + kp (16 tiles) + vp (16 tiles) = 40 jobs over 8 waves
    for (int job = wave; job < 40; job += 8) {
      int mat, rt, nt;
      if (job < 8) { mat = 0; rt = 0; nt = job; }
      else { int j = job - 8; mat = 1 + (j >> 4); j &= 15; rt = j >> 3; nt = j & 7; }
      const int n  = nt * 16 + lm;
      const int nc = (n < OFC) ? n : (OFC - 1);          // clamp: bulk loop stays guard-free
      const float* wrow = in_w + mat * OFC * OFC + nc * OFC;
      const float bias = (n < OFC) ? in_b[mat * OFC + n] : 0.f;
      // A base: eeg row (mat==0) or sliding-window key row offset
      const int abase = (mat == 0) ? ((1 + lm) * 140 + 22) : (xrow * 140 + rt * 16 + lm);
      v8f c = {0.f,0.f,0.f,0.f,0.f,0.f,0.f,0.f};
      for (int k0 = 0; k0 < 116; k0 += 4) {              // k <= 115 : all valid, no guards
        const int k = k0 + koff;
        v2f a, b;
        a.x = xbuf[abase + k];  a.y = xbuf[abase + k + 1];
        b.x = wrow[k];          b.y = wrow[k + 1];
        c = wmma4(a, b, c);
      }
      { // peeled tail k = 116..119 (only place K-guards are needed)
        const int k = 116 + koff;
        v2f a, b;
        a.x = xbuf[abase + k];  a.y = xbuf[abase + k + 1];   // don't-care beyond 117: b is 0
        b.x = (n < OFC && k     < OFC) ? wrow[k]     : 0.f;
        b.y = (n < OFC && k + 1 < OFC) ? wrow[k + 1] : 0.f;
        c = wmma4(a, b, c);
      }
      for (int r = 0; r < 8; ++r) {
        const int row = r + half * 8;
        const float v = (n < OFC) ? (c[r] + bias) : 0.f;   // keep pad columns exactly 0
        if (mat == 0) U.td.qp[row][n] = v * RSQ;
        else {
          const int g = rt * 16 + row;
          if (g < TDN) { if (mat == 1) U.td.kp[g][n] = v; else U.td.vp[g][n] = v; }
        }
      }
    }
    __syncthreads();

    // scores = qp @ kp^T  (16x21, 2 column tiles) -- all-LDS, guard-free
    if (wave < 2) {
      const int n = wave * 16 + lm;
      v8f c = {0.f,0.f,0.f,0.f,0.f,0.f,0.f,0.f};
      for (int k0 = 0; k0 < KPAD; k0 += 4) {
        const int k = k0 + koff;
        v2f a, b;
        a.x = U.td.qp[lm][k]; a.y = U.td.qp[lm][k + 1];
        b.x = U.td.kp[n][k];  b.y = U.td.kp[n][k + 1];
        c = wmma4(a, b, c);
      }
      for (int r = 0; r < 8; ++r) U.td.sc[r + half * 8][n] = c[r];
    }
    __syncthreads();

    if (tid < 16) { // row softmax over 21 keys
      float m = -1e30f;
      for (int j = 0; j < TDN; ++j) m = fmaxf(m, U.td.sc[tid][j]);
      float s = 0.f;
      for (int j = 0; j < TDN; ++j) { float e = __expf(U.td.sc[tid][j] - m); U.td.sc[tid][j] = e; s += e; }
      const float inv = 1.f / s;
      for (int j = 0; j < TDN; ++j) U.td.sc[tid][j] *= inv;
    }
    __syncthreads();

    { // attnV = attn @ vp  (K=21 pad 24) -> reuse qp buffer (vp pads zero => pads stay 0)
      const int n = wave * 16 + lm;
      v8f c = {0.f,0.f,0.f,0.f,0.f,0.f,0.f,0.f};
      for (int k0 = 0; k0 < 24; k0 += 4) {
        const int k = k0 + koff;
        v2f a, b;
        a.x = U.td.sc[lm][k]; a.y = U.td.sc[lm][k + 1];
        b.x = U.td.vp[k][n];  b.y = U.td.vp[k + 1][n];
        c = wmma4(a, b, c);
      }
      for (int r = 0; r < 8; ++r) U.td.qp[r + half * 8][n] = c[r];
    }
    __syncthreads();

    { // att = attnV @ out_w^T + out_b  (clamped B + peeled tail)
      const int n  = wave * 16 + lm;
      const int nc = (n < OFC) ? n : (OFC - 1);
      const float* wrow = ow + nc * OFC;
      v8f c = {0.f,0.f,0.f,0.f,0.f,0.f,0.f,0.f};
      for (int k0 = 0; k0 < 116; k0 += 4) {
        const int k = k0 + koff;
        v2f a, b;
        a.x = U.td.qp[lm][k]; a.y = U.td.qp[lm][k + 1];
        b.x = wrow[k];        b.y = wrow[k + 1];
        c = wmma4(a, b, c);
      }
      {
        const int k = 116 + koff;
        v2f a, b;
        a.x = U.td.qp[lm][k]; a.y = U.td.qp[lm][k + 1];
        b.x = (n < OFC && k     < OFC) ? wrow[k]     : 0.f;
        b.y = (n < OFC && k + 1 < OFC) ? wrow[k + 1] : 0.f;
        c = wmma4(a, b, c);
      }
      const float bias = (n < OFC) ? ob[n] : 0.f;
      for (int r = 0; r < 8; ++r)
        U.td.att[r + half * 8][n] = (n < OFC) ? (c[r] + bias) : 0.f;
    }
    __syncthreads();

    // argmax_i  eeg[i] . (column-sum over rows of att)
    if (tid < 128) {
      float s = 0.f;
      for (int i = 0; i < 16; ++i) s += U.td.att[i][tid];
      gsum[tid] = s;
    }
    __syncthreads();
    if (tid < 16) {
      float s = 0.f;
      for (int k = 0; k < OFC; ++k) s += xbuf[(1 + tid) * 140 + 22 + k] * gsum[k];
      sval[tid] = s;
    }
    __syncthreads();
    if (tid == 0) {
      int best = 0; float bm = sval[0];
      for (int i = 1; i < 16; ++i) if (sval[i] > bm) { bm = sval[i]; best = i; }
      iSel = best;
    }
    __syncthreads();
    {
      float* arow = sel ? attrowB : attrowA;
      if (tid < OFC) arow[tid] = U.td.att[iSel][tid];
    }
    __syncthreads();
  }

  // ================= cross-modal branches =================
  for (int i = tid; i < 160; i += 256) {   // re-zero branch pad rows
    const int r = 118 + i / 16, cc = i & 15;
    U.br.qp[r][cc] = 0.f; U.br.kp[r][cc] = 0.f; U.br.vp[r][cc] = 0.f; U.br.o[r][cc] = 0.f;
  }
  __syncthreads();

  for (int brn = 0; brn < 4; ++brn) {
    // data = [wA, eeg, eeg, wB]; kv = [eeg, wA, wB, eeg]; src: 0=wA 1=eeg^T 2=wB
    const int dsrc = (brn == 0) ? 0 : (brn == 3) ? 2 : 1;
    const int ksrc = (brn == 1) ? 0 : (brn == 2) ? 2 : 1;

    // qkv projections: 3 mats x 8 row tiles = 24 jobs (A guard-free: bad rows discarded)
    for (int job = wave; job < 24; job += 8) {
      const int mat = job >> 3, rt = job & 7;
      const int src = (mat == 0) ? dsrc : ksrc;
      const float* wmat = cm_in_w + (brn * 48 + mat * 16) * 16;
      const int n = lm;
      v8f c = {0.f,0.f,0.f,0.f,0.f,0.f,0.f,0.f};
      for (int k0 = 0; k0 < 16; k0 += 4) {
        const int k = k0 + koff;
        const int j = rt * 16 + lm;
        v2f a, b;
        if (src == 1) { a.x = xbuf[(1 + k) * 140 + 22 + j]; a.y = xbuf[(2 + k) * 140 + 22 + j]; }
        else if (src == 0) { a.x = projA[k] * attrowA[j]; a.y = projA[k + 1] * attrowA[j]; }
        else { a.x = projB[k] * attrowB[j]; a.y = projB[k + 1] * attrowB[j]; }
        b.x = wmat[n * 16 + k]; b.y = wmat[n * 16 + k + 1];
        c = wmma4(a, b, c);
      }
      const float bias = cm_in_b[brn * 48 + mat * 16 + n];
      const float scl = (mat == 0) ? 0.25f : 1.0f; // 1/sqrt(16) on q
      float (*dst)[16] = (mat == 0) ? U.br.qp : (mat == 1) ? U.br.kp : U.br.vp;
      for (int r = 0; r < 8; ++r) {
        const int row = rt * 16 + r + half * 8;
        if (row < OFC) dst[row][n] = (c[r] + bias) * scl;
      }
    }
    __syncthreads();

    // flash-style attention: one 16-row score strip at a time
    for (int rt = 0; rt < 8; ++rt) {
      { // scores strip: each wave one 16-col tile
        const int n = wave * 16 + lm;
        v8f c = {0.f,0.f,0.f,0.f,0.f,0.f,0.f,0.f};
        for (int k0 = 0; k0 < 16; k0 += 4) {
          const int k = k0 + koff;
          v2f a, b;
          a.x = U.br.qp[rt * 16 + lm][k]; a.y = U.br.qp[rt * 16 + lm][k + 1];
          b.x = U.br.kp[n][k];            b.y = U.br.kp[n][k + 1];
          c = wmma4(a, b, c);
        }
        for (int r = 0; r < 8; ++r) U.br.sc[r + half * 8][n] = c[r];
      }
      __syncthreads();
      if (tid < 16 && rt * 16 + tid < OFC) { // softmax over 118 valid cols
        float m = -1e30f;
        for (int j = 0; j < OFC; ++j) m = fmaxf(m, U.br.sc[tid][j]);
        float s = 0.f;
        for (int j = 0; j < OFC; ++j) { float e = __expf(U.br.sc[tid][j] - m); U.br.sc[tid][j] = e; s += e; }
        const float inv = 1.f / s;
        for (int j = 0; j < OFC; ++j) U.br.sc[tid][j] *= inv;
      }
      __syncthreads();
      if (wave == 0) { // attn@V (K=118) then output projection (K=16)
        v8f c = {0.f,0.f,0.f,0.f,0.f,0.f,0.f,0.f};
        for (int k0 = 0; k0 < KPAD; k0 += 4) {
          const int k = k0 + koff;
          v2f a, b;
          a.x = U.br.sc[lm][k]; a.y = U.br.sc[lm][k + 1];
          b.x = U.br.vp[k][lm]; b.y = U.br.vp[k + 1][lm];
          c = wmma4(a, b, c);
        }
        for (int r = 0; r < 8; ++r) U.br.tmp[r + half * 8][lm] = c[r];
        v8f c2 = {0.f,0.f,0.f,0.f,0.f,0.f,0.f,0.f};
        for (int k0 = 0; k0 < 16; k0 += 4) {
          const int k = k0 + koff;
          v2f a, b;
          a.x = U.br.tmp[lm][k]; a.y = U.br.tmp[lm][k + 1];
          b.x = cm_out_w[(brn * 16 + lm) * 16 + k];
          b.y = cm_out_w[(brn * 16 + lm) * 16 + k + 1];
          c2 = wmma4(a, b, c2);
        }
        const float bias = cm_out_b[brn * 16 + lm];
        for (int r = 0; r < 8; ++r) {
          const int row = rt * 16 + r + half * 8;
          if (row < OFC) U.br.o[row][lm] = c2[r] + bias;
        }
      }
      __syncthreads();
    }

    // conv1d (10 ch, K=9 over 16 lanes) + relu + global max
    {
      float* ybuf = &U.br.sc[0][0]; // sc strip is free now
      for (int j = tid; j < 1100; j += 256) {
        const int ch = j / 110, t = j % 110;
        const float* w = conv_w + (brn * 10 + ch) * 144;
        float s = conv_b[brn * 10 + ch];
        for (int cc = 0; cc < 16; ++cc) {
          const float* wr = w + cc * 9;
          for (int q = 0; q < 9; ++q) s += wr[q] * U.br.o[t + q][cc];
        }
        ybuf[ch * 110 + t] = fmaxf(s, 0.f);
      }
      __syncthreads();
      if (tid < 10) {
        float m = ybuf[tid * 110];
        for (int t = 1; t < 110; ++t) m = fmaxf(m, ybuf[tid * 110 + t]);
        feats[brn * 10 + tid] = m;
      }
      __syncthreads();
    }
  }

  // ================= classifier head =================
  if (tid < 40) {
    float s = fc1_b[tid];
    for (int v = 0; v < 40; ++v) s += feats[v] * fc1_w[tid * 40 + v];
    hbuf[tid] = 1.f / (1.f + __expf(-s));
  }
  __syncthreads();
  if (tid < 2) {
    float s = fc2_b[tid];
    for (int u = 0; u < 40; ++u) s += hbuf[u] * fc2_w[tid * 40 + u];
    out[tid] = 1.f / (1.f + __expf(-s));
  }
}

extern "C" void kernel_launch(void* const* d_in, const int* in_sizes, int n_in,
                              void* d_out, int out_size, void* d_ws, size_t ws_size,
                              hipStream_t stream) {
  (void)in_sizes; (void)n_in; (void)out_size; (void)d_ws; (void)ws_size;
  fused_net<<<1, 256, 0, stream>>>(
      (const float*)d_in[0],
      (const float*)d_in[1],  (const float*)d_in[2],
      (const float*)d_in[3],  (const float*)d_in[4],
      (const float*)d_in[5],  (const float*)d_in[6],
      (const float*)d_in[7],  (const float*)d_in[8],
      (const float*)d_in[9],  (const float*)d_in[10],
      (const float*)d_in[11], (const float*)d_in[12],
      (const float*)d_in[13], (const float*)d_in[14],
      (const float*)d_in[15], (const float*)d_in[16],
      (const float*)d_in[17], (const float*)d_in[18],
      (const float*)d_in[19], (const float*)d_in[20],
      (float*)d_out);
}